// SpatiallyVaryingAnisotropicSimulator_14173392077033
// MI455X (gfx1250) — compile-verified
//
#include <hip/hip_runtime.h>
#include <cstdint>

// ---------------------------------------------------------------------------
// 2-D anisotropic elastic wave propagator for MI455X (gfx1250).
//  - 192 sequential steps as 192 stream launches (graph-captured by harness)
//  - padded state arrays in d_ws (zero ring == fixed boundary)
//  - 32x32 tiles staged into LDS via CDNA5 async global->LDS B128 copies
//    (tracked with ASYNCcnt / s_wait_asynccnt)
// ---------------------------------------------------------------------------

#define NXX 384
#define NYY 384
#define PADP 388           // padded row pitch (>= NYY+2, multiple of 4 floats)
#define PADR 386           // padded rows (NXX+2)
#define NTT 192
#define TILE 32
#define LPITCH 36          // LDS row pitch (34 used + 2 spare, 144B = 9x16B)

__device__ __constant__ const float kHinv2 = 1.0e8f;          // 1/H^2, H=1e-4
#define DT2f     2.5e-17f                                      // (5e-9)^2
#define RHOINVf  (1.0f / 1610.0f)

#if __has_builtin(__builtin_amdgcn_global_load_async_to_lds_b128)
#define ASYNC128 1
#else
#define ASYNC128 0
#endif
#if __has_builtin(__builtin_amdgcn_global_load_async_to_lds_b32)
#define ASYNC32 1
#else
#define ASYNC32 0
#endif

typedef int v4i_t __attribute__((vector_size(16)));            // 4 x i32
typedef __attribute__((address_space(1))) int   gint_t;        // global i32
typedef __attribute__((address_space(3))) int   lint_t;        // LDS i32
typedef __attribute__((address_space(1))) v4i_t gv4i_t;        // global v4i32
typedef __attribute__((address_space(3))) v4i_t lv4i_t;        // LDS v4i32

__device__ __forceinline__ void async_wait0() {
#if __has_builtin(__builtin_amdgcn_s_wait_asynccnt)
  __builtin_amdgcn_s_wait_asynccnt(0);
#else
  asm volatile("s_wait_asynccnt 0" ::: "memory");
#endif
}

__device__ __forceinline__ void async_cp_b128(const float* g, float* l) {
#if ASYNC128
  __builtin_amdgcn_global_load_async_to_lds_b128(
      (gv4i_t*)(uintptr_t)g, (lv4i_t*)(uintptr_t)l, 0, 0);
#elif ASYNC32
#pragma unroll
  for (int q = 0; q < 4; ++q)
    __builtin_amdgcn_global_load_async_to_lds_b32(
        (gint_t*)(uintptr_t)(g + q), (lint_t*)(uintptr_t)(l + q), 0, 0);
#else
#pragma unroll
  for (int q = 0; q < 4; ++q) l[q] = g[q];
#endif
}

__device__ __forceinline__ float clampf(float v, float lo, float hi) {
  return fminf(fmaxf(v, lo), hi);
}

// ---------------------------------------------------------------------------
// Init: clip(exp(log_C)) coefficient fields + zero padded displacement bufs
// ---------------------------------------------------------------------------
__global__ __launch_bounds__(256) void init_kernel(
    const float* __restrict__ lC11, const float* __restrict__ lC12,
    const float* __restrict__ lC16, const float* __restrict__ lC22,
    const float* __restrict__ lC26, const float* __restrict__ lC66,
    float* __restrict__ X0, float* __restrict__ X1,
    float* __restrict__ Y0, float* __restrict__ Y1,
    float* __restrict__ C11, float* __restrict__ C22,
    float* __restrict__ C66, float* __restrict__ C16,
    float* __restrict__ C26, float* __restrict__ C12p66) {
  const int i = blockIdx.x * blockDim.x + threadIdx.x;
  if (i < PADR * PADP) {
    X0[i] = 0.0f; X1[i] = 0.0f; Y0[i] = 0.0f; Y1[i] = 0.0f;
  }
  if (i < NXX * NYY) {
    const float c11 = clampf(__expf(lC11[i]), 5.0e10f, 2.5e11f);
    const float c22 = clampf(__expf(lC22[i]), 5.0e9f,  5.0e10f);
    const float c12 = clampf(__expf(lC12[i]), 5.0e9f,  5.0e10f);
    const float c16 = clampf(__expf(lC16[i]), 0.0f,    6.0e10f);
    const float c26 = clampf(__expf(lC26[i]), 0.0f,    2.0e10f);
    const float c66 = clampf(__expf(lC66[i]), 5.0e9f,  3.0e10f);
    C11[i] = c11; C22[i] = c22; C66[i] = c66;
    C16[i] = c16; C26[i] = c26; C12p66[i] = c12 + c66;
  }
}

// ---------------------------------------------------------------------------
// One leap-frog step.  u1* = current field (padded), u2* = previous field
// (padded) which is overwritten in place with the new field.
// ---------------------------------------------------------------------------
__global__ __launch_bounds__(256) void step_kernel(
    const float* __restrict__ u1x, const float* __restrict__ u1y,
    float* __restrict__ u2x, float* __restrict__ u2y,
    const float* __restrict__ C11, const float* __restrict__ C22,
    const float* __restrict__ C66, const float* __restrict__ C16,
    const float* __restrict__ C26, const float* __restrict__ C12p66,
    const float* __restrict__ gauss, const float* __restrict__ sig, int t,
    float* __restrict__ outx, float* __restrict__ outy) {
  __shared__ __align__(16) float sx[34 * LPITCH];
  __shared__ __align__(16) float sy[34 * LPITCH];

  const int tid  = threadIdx.x;
  const int row0 = blockIdx.x * TILE;  // halo-tile origin in padded coords
  const int col0 = blockIdx.y * TILE;

  // Stage 34x36 halo tiles of u1x / u1y into LDS with async B128 DMA:
  // 34 rows x 9 xfers of 4 floats each (16B-aligned in both global and LDS).
  for (int i = tid; i < 34 * 9; i += 256) {
    const int r = i / 9, j4 = (i % 9) * 4;
    const int g = (row0 + r) * PADP + col0 + j4;
    const int l = r * LPITCH + j4;
    async_cp_b128(u1x + g, sx + l);
    async_cp_b128(u1y + g, sy + l);
  }
  async_wait0();
  __syncthreads();

  const float st = sig[t];        // wave-uniform scalar (SMEM broadcast)
  const int tx = tid & 31;        // column within tile
  const int ty = tid >> 5;        // base row (0..7); each thread does 4 rows

#pragma unroll
  for (int k = 0; k < 4; ++k) {
    const int lr = 1 + ty + 8 * k;
    const int lc = 1 + tx;
    const int b  = lr * LPITCH + lc;

    const float x_mm = sx[b - LPITCH - 1], x_m0 = sx[b - LPITCH], x_mp = sx[b - LPITCH + 1];
    const float x_0m = sx[b - 1],          x_00 = sx[b],          x_0p = sx[b + 1];
    const float x_pm = sx[b + LPITCH - 1], x_p0 = sx[b + LPITCH], x_pp = sx[b + LPITCH + 1];

    const float y_mm = sy[b - LPITCH - 1], y_m0 = sy[b - LPITCH], y_mp = sy[b - LPITCH + 1];
    const float y_0m = sy[b - 1],          y_00 = sy[b],          y_0p = sy[b + 1];
    const float y_pm = sy[b + LPITCH - 1], y_p0 = sy[b + LPITCH], y_pp = sy[b + LPITCH + 1];

    const float d2x_x = (x_m0 - 2.0f * x_00 + x_p0) * kHinv2;
    const float d2y_x = (x_0m - 2.0f * x_00 + x_0p) * kHinv2;
    const float dxy_x = 0.25f * (x_mm - x_mp - x_pm + x_pp) * kHinv2;
    const float d2x_y = (y_m0 - 2.0f * y_00 + y_p0) * kHinv2;
    const float d2y_y = (y_0m - 2.0f * y_00 + y_0p) * kHinv2;
    const float dxy_y = 0.25f * (y_mm - y_mp - y_pm + y_pp) * kHinv2;

    const int gr = row0 + ty + 8 * k;      // interior cell coords (0..383)
    const int gc = col0 + tx;
    const int ci = gr * NYY + gc;          // unpadded index (coeffs/output)
    const int pi = (gr + 1) * PADP + (gc + 1);  // padded index (state)

    const float c11 = C11[ci], c22 = C22[ci], c66 = C66[ci];
    const float c16 = C16[ci], c26 = C26[ci], cps = C12p66[ci];

    const float lux = c11 * d2x_x + c66 * d2y_x + cps * dxy_y +
                      2.0f * c16 * dxy_x + c16 * d2x_y + c26 * d2y_y;
    const float luy = c66 * d2x_y + c22 * d2y_y + cps * dxy_x +
                      c16 * d2x_x + c26 * d2y_x + 2.0f * c26 * dxy_y;

    const float ax = lux * RHOINVf;
    const float ay = (luy + st * gauss[ci]) * RHOINVf;

    const float unx = 2.0f * x_00 - u2x[pi] + DT2f * ax;
    const float uny = 2.0f * y_00 - u2y[pi] + DT2f * ay;
    u2x[pi] = unx;
    u2y[pi] = uny;
    if (outx != nullptr) {
      outx[ci] = unx;
      outy[ci] = uny;
    }
  }
}

// ---------------------------------------------------------------------------
extern "C" void kernel_launch(void* const* d_in, const int* in_sizes, int n_in,
                              void* d_out, int out_size, void* d_ws,
                              size_t ws_size, hipStream_t stream) {
  (void)in_sizes; (void)n_in; (void)out_size; (void)ws_size;
  const float* lC11 = (const float*)d_in[0];
  const float* lC12 = (const float*)d_in[1];
  const float* lC16 = (const float*)d_in[2];
  const float* lC22 = (const float*)d_in[3];
  const float* lC26 = (const float*)d_in[4];
  const float* lC66 = (const float*)d_in[5];
  const float* gauss = (const float*)d_in[6];
  const float* sig   = (const float*)d_in[7];
  float* out = (float*)d_out;
  float* ws  = (float*)d_ws;

  const size_t PP = (size_t)PADR * PADP;   // padded field size
  const size_t NN = (size_t)NXX * NYY;     // interior field size

  float* X0 = ws;           float* X1 = X0 + PP;
  float* Y0 = X1 + PP;      float* Y1 = Y0 + PP;
  float* C11 = Y1 + PP;     float* C22 = C11 + NN;
  float* C66 = C22 + NN;    float* C16 = C66 + NN;
  float* C26 = C16 + NN;    float* C12p66 = C26 + NN;

  const int initBlocks = (int)((PP + 255) / 256);
  init_kernel<<<initBlocks, 256, 0, stream>>>(lC11, lC12, lC16, lC22, lC26,
                                              lC66, X0, X1, Y0, Y1, C11, C22,
                                              C66, C16, C26, C12p66);

  float* X[2] = {X0, X1};
  float* Y[2] = {Y0, Y1};
  const dim3 grid(NXX / TILE, NYY / TILE);  // 12 x 12 = 144 blocks

  for (int t = 0; t < NTT; ++t) {
    float* ox = nullptr;
    float* oy = nullptr;
    if ((t & 3) == 3) {
      const size_t s = (size_t)(t >> 2);
      ox = out + s * NN;               // [0, 0, s, :, :]
      oy = out + (48 + s) * NN;        // [1, 0, s, :, :]
    }
    step_kernel<<<grid, 256, 0, stream>>>(
        X[t & 1], Y[t & 1], X[(t + 1) & 1], Y[(t + 1) & 1], C11, C22, C66, C16,
        C26, C12p66, gauss, sig, t, ox, oy);
  }
}